// DTW_BOOSTING_38448547233970
// MI455X (gfx1250) — compile-verified
//
#include <hip/hip_runtime.h>

typedef __attribute__((ext_vector_type(2))) float v2f;
typedef __attribute__((ext_vector_type(8))) float v8f;

#define KLEN   512
#define NLEN   65536
#define CHUNK  16
#define DPITCH (KLEN + 1)          // pad to keep LDS banks conflict-free on tile store
#define INF_F  __builtin_inff()

// ---------------------------------------------------------------------------
// Forward DTW: one persistent workgroup (512 threads = 16 wave32) sweeps the
// 65536 columns. Per 16-column panel, one f32 WMMA per 16x16 tile computes
// the complete local distance d = (k-x)^2 (A = -2k, B = x, C seeded k^2+x^2).
// The min-plus column recurrence is an associative scan: 5-stage intra-wave
// shuffle scan + 4-stage lane-parallel scan of the 16 wave aggregates.
// Left/diag neighbors are register carries and move codes are packed 2 bits
// per cell into one uint32 per thread per chunk, so the serial per-column
// chain is: scan + one dscnt-only barrier — no global store on it.
// Packed move words (8 MiB) live in d_ws, resident in the 192 MB L2.
// ---------------------------------------------------------------------------
__global__ __launch_bounds__(KLEN) void dtw_forward(
    const float* __restrict__ x, const float* __restrict__ kern,
    unsigned int* __restrict__ movesW)
{
  __shared__ float s_k  [KLEN];
  __shared__ float s_k2 [KLEN];
  __shared__ float s_x  [CHUNK];
  __shared__ float s_x2 [CHUNK];
  __shared__ float s_d  [CHUNK * DPITCH];   // d panel, layout [n][row]
  __shared__ float s_aggS[2][16];           // parity-buffered wave aggregates
  __shared__ float s_aggM[2][16];

  const int tid  = (int)threadIdx.x;
  const int lane = tid & 31;
  const int wave = tid >> 5;
  const int rb   = wave * 32;               // this wave's 32 rows

  {
    float kv = kern[tid];
    s_k[tid]  = kv;
    s_k2[tid] = kv * kv;
  }
  __syncthreads();

  // A operands: 16x4 f32, only K=0 slot nonzero, value -2*k (full-precision)
  v2f a0, a1;
  a0.x = (lane < 16) ? -2.f * s_k[rb + lane]      : 0.f;  a0.y = 0.f;
  a1.x = (lane < 16) ? -2.f * s_k[rb + 16 + lane] : 0.f;  a1.y = 0.f;

  // loop-invariant k^2 values in this lane's C/D slots
  const int n  = lane & 15;
  const int mo = (lane < 16) ? 0 : 8;
  float k2r0[8], k2r1[8];
  #pragma unroll
  for (int v = 0; v < 8; ++v) {
    k2r0[v] = s_k2[rb + v + mo];
    k2r1[v] = s_k2[rb + 16 + v + mo];
  }

  float left_c = INF_F;                     // cost[tid][j-1]   (register carry)
  float diag_c = INF_F;                     // cost[tid-1][j-1] (register carry)

  for (int j0 = 0; j0 < NLEN; j0 += CHUNK) {
    if (tid < CHUNK) { float xv = x[j0 + tid]; s_x[tid] = xv; s_x2[tid] = xv * xv; }
    __syncthreads();

    // B operand: 4x16 f32, only K=0 row nonzero; C seeded with k^2 + x^2
    v2f bm; bm.x = (lane < 16) ? s_x[lane] : 0.f;  bm.y = 0.f;
    float xx = s_x2[n];
    v8f c0, c1;
    #pragma unroll
    for (int v = 0; v < 8; ++v) { c0[v] = k2r0[v] + xx; c1[v] = k2r1[v] + xx; }

    // d = (-2k)*x + (k^2 + x^2) = (k - x)^2, straight off the matrix pipe
    c0 = __builtin_amdgcn_wmma_f32_16x16x4_f32(false, a0, false, bm, (short)0, c0, false, false);
    c1 = __builtin_amdgcn_wmma_f32_16x16x4_f32(false, a1, false, bm, (short)0, c1, false, false);

    #pragma unroll
    for (int v = 0; v < 8; ++v) {
      s_d[n * DPITCH + (rb + v + mo)]      = c0[v];
      s_d[n * DPITCH + (rb + 16 + v + mo)] = c1[v];
    }
    __syncthreads();

    unsigned int packed = 0u;               // 16 x 2-bit move codes

    for (int jj = 0; jj < CHUNK; ++jj) {
      const int j = j0 + jj;
      const int p = j & 1;

      float d  = s_d[jj * DPITCH + tid];
      float pv = left_c;                                        // left
      float dg = (tid == 0) ? ((j == 0) ? 0.f : INF_F)
                            : diag_c;                           // diag
      // scan element: (s, m), combine((s1,m1),(s2,m2)) = (s1+s2, min(m1+s2, m2))
      float s = d;
      float m = d + fminf(pv, dg);

      #pragma unroll
      for (int off = 1; off < 32; off <<= 1) {
        float s2 = __shfl_up(s, off, 32);
        float m2 = __shfl_up(m, off, 32);
        if (lane >= off) { m = fminf(m2 + s, m); s = s2 + s; }
      }
      if (lane == 31) { s_aggS[p][wave] = s; s_aggM[p][wave] = m; }
      __syncthreads();

      // lane-parallel scan of the 16 wave aggregates (mirrored in both halves)
      const int hidx = lane & 15;
      float as = s_aggS[p][hidx];
      float am = s_aggM[p][hidx];
      #pragma unroll
      for (int off = 1; off < 16; off <<= 1) {
        float as2 = __shfl_up(as, off, 32);
        float am2 = __shfl_up(am, off, 32);
        if (hidx >= off) { am = fminf(am2 + as, am); as = as2 + as; }
      }
      // exclusive prefix for this wave = inclusive aggregate at index wave-1
      float Mw = __shfl(am, wave - 1, 32);
      float M  = (wave == 0) ? INF_F : Mw;

      float col = fminf(M + s, m);          // cost[tid][j]

      // up neighbor: shuffle within wave; wave-boundary value == prefix M
      float upn  = __shfl_up(col, 1, 32);
      float c_up = (lane == 0) ? ((wave == 0) ? INF_F : M) : upn;

      // argmin over (diag, up, left) with jnp.argmin tie order
      int mv = 0; float best = dg;
      if (c_up < best) { best = c_up; mv = 1; }
      if (pv   < best) {              mv = 2; }
      packed |= (unsigned int)mv << (jj * 2);

      left_c = col;
      diag_c = c_up;
    }

    // one coalesced b32 store per chunk (2 KB per workgroup)
    movesW[(unsigned int)(j0 >> 4) * KLEN + tid] = packed;
  }
}

// ---------------------------------------------------------------------------
// Backtrack walker: single thread replays 2-bit move codes from (511, 65535)
// down to (0,0). Path visits every column; out[j] is flushed exactly once
// when j decrements. loss = path SSE, stored at d_out[65536].
// ---------------------------------------------------------------------------
__global__ void dtw_backtrack(
    const float* __restrict__ x, const float* __restrict__ kern,
    const unsigned int* __restrict__ movesW, float* __restrict__ out)
{
  if (threadIdx.x != 0 || blockIdx.x != 0) return;

  int i = KLEN - 1, j = NLEN - 1;
  float loss = 0.f, acc = 0.f;
  while (true) {
    float kv = kern[i];
    float xv = x[j];
    float df = kv - xv;
    loss += df * df;
    acc  += kv;
    if (i == 0 && j == 0) { out[0] = (xv - acc) * 0.5f; break; }
    unsigned int w = movesW[(unsigned int)(j >> 4) * KLEN + i];
    int mv = (int)((w >> ((j & 15) * 2)) & 3u);
    if (j >= 16)
      __builtin_prefetch(&movesW[(unsigned int)((j - 16) >> 4) * KLEN + i], 0, 1);
    if (mv != 1) { out[j] = (xv - acc) * 0.5f; acc = 0.f; }   // j will decrement
    i -= (mv != 2);
    j -= (mv != 1);
  }
  out[NLEN] = loss;
}

extern "C" void kernel_launch(void* const* d_in, const int* in_sizes, int n_in,
                              void* d_out, int out_size, void* d_ws, size_t ws_size,
                              hipStream_t stream) {
  const float* x    = (const float*)d_in[0];   // [65536]
  const float* kern = (const float*)d_in[1];   // [512]
  float* out = (float*)d_out;                  // [65537]: out ++ loss
  unsigned int* movesW = (unsigned int*)d_ws;  // needs (65536/16)*512*4 = 8 MiB

  dtw_forward  <<<1, KLEN, 0, stream>>>(x, kern, movesW);
  dtw_backtrack<<<1, 32,   0, stream>>>(x, kern, movesW, out);
}